// RandomSample_34608846471730
// MI455X (gfx1250) — compile-verified
//
#include <hip/hip_runtime.h>

typedef __attribute__((ext_vector_type(2))) float v2f;
typedef __attribute__((ext_vector_type(8))) float v8f;

// Problem constants (from reference): B=4, N=16384, DIMS=3, F_DIMS=64
constexpr int BATCH  = 4;
constexpr int NPTS   = 16384;
constexpr int MHALF  = 8192;          // N/2 valid + N/2 invalid
constexpr int FD4    = 16;            // 64 floats = 16 float4 per feature row

// Flat output layout (floats), in reference return order
constexpr size_t VPC_OFF  = 0;                               // valid_pc  (B,8192,3)
constexpr size_t VPC_SZ   = (size_t)BATCH * MHALF * 3;       // 98304
constexpr size_t VF_OFF   = VPC_OFF + VPC_SZ;                // valid_feats (B,8192,64)
constexpr size_t VF_SZ    = (size_t)BATCH * MHALF * 64;      // 2097152
constexpr size_t NIDX_OFF = VF_OFF + VF_SZ;                  // n_idx (B,8192)
constexpr size_t NIDX_SZ  = (size_t)BATCH * MHALF;           // 32768
constexpr size_t RNDS_OFF = NIDX_OFF + NIDX_SZ;              // rnds (1,16384)

// ---------------------------------------------------------------------------
// Kernel 1: gather valid_pc to output, build WMMA-ready point tables in ws,
//           and emit rnds = (float)perm.
//   qpts[b][i] = (-2x, -2y, -2z, 1)        from pc[b][perm[8192+i]]  (queries)
//   ppts[b][i] = ( x,   y,   z,  x2+y2+z2) from pc[b][perm[i]]       (targets)
// ---------------------------------------------------------------------------
__global__ __launch_bounds__(256) void gather_pts_kernel(
    const float* __restrict__ pc, const int* __restrict__ perm,
    float* __restrict__ out, float4* __restrict__ ppts, float4* __restrict__ qpts)
{
  int t = blockIdx.x * blockDim.x + threadIdx.x;   // 0 .. B*8192-1
  int b = t >> 13;
  int i = t & (MHALF - 1);
  int vi = perm[i];
  int qi = perm[MHALF + i];
  const float* pcb = pc + (size_t)b * NPTS * 3;

  float x = pcb[vi * 3 + 0];
  float y = pcb[vi * 3 + 1];
  float z = pcb[vi * 3 + 2];
  size_t o = (size_t)b * MHALF + i;
  out[VPC_OFF + o * 3 + 0] = x;
  out[VPC_OFF + o * 3 + 1] = y;
  out[VPC_OFF + o * 3 + 2] = z;
  ppts[o] = make_float4(x, y, z, x * x + y * y + z * z);

  float qx = pcb[qi * 3 + 0];
  float qy = pcb[qi * 3 + 1];
  float qz = pcb[qi * 3 + 2];
  qpts[o] = make_float4(-2.0f * qx, -2.0f * qy, -2.0f * qz, 1.0f);

  if (t < NPTS) out[RNDS_OFF + t] = (float)perm[t];
}

// ---------------------------------------------------------------------------
// Kernel 2: gather valid_feats with float4 vector moves (coalesced).
// ---------------------------------------------------------------------------
__global__ __launch_bounds__(256) void gather_feats_kernel(
    const float4* __restrict__ feats4, const int* __restrict__ perm,
    float4* __restrict__ out4)
{
  int t  = blockIdx.x * blockDim.x + threadIdx.x;  // 0 .. B*8192*16-1
  int b  = t >> 17;                                // 8192*16 = 2^17
  int r  = t & ((MHALF * FD4) - 1);
  int i  = r >> 4;
  int f4 = r & (FD4 - 1);
  int vi = perm[i];
  out4[((size_t)b * MHALF + i) * FD4 + f4] =
      feats4[((size_t)b * NPTS + vi) * FD4 + f4];
}

// ---------------------------------------------------------------------------
// Kernel 3: WMMA-tiled 2nd-nearest-neighbor search.
// One wave32 owns 16 queries of one batch. Per 16-target chunk:
//   S = A(16x4) * B(4x16) via V_WMMA_F32_16X16X4_F32,
//   S[q][p] = -2 q.p + |p|^2   (row-constant |q|^2 dropped: argmin-invariant)
// then per-lane streaming top-2 update; final cross-lane butterfly merge.
//
// Operand layouts (CDNA5 ISA 7.12.2, wave32):
//   A 16x4 : lane L -> M=L&15 ; VGPR j -> K = j + 2*(L>=16)
//   B 4x16 : lane L -> N=L&15 ; VGPR j -> K = j + 2*(L>=16)
//   C 16x16: VGPR s, lane L -> M = s + 8*(L>=16), N = L&15
// ---------------------------------------------------------------------------
__global__ __launch_bounds__(256) void knn2_wmma_kernel(
    const float4* __restrict__ qpts, const float4* __restrict__ ppts,
    float* __restrict__ nidx_out)
{
  const int tid   = blockIdx.x * blockDim.x + threadIdx.x;
  const int wave  = tid >> 5;                 // 0 .. 2047
  const int lane  = threadIdx.x & 31;
  const int b     = wave >> 9;                // 512 waves per batch
  const int qBase = (wave & 511) << 4;        // 16 queries per wave
  const int hi    = lane >> 4;                // upper lane half?
  const int col   = lane & 15;

  const float4* qp = qpts + (size_t)b * MHALF;
  const float4* pp = ppts + (size_t)b * MHALF;

  // Loop-invariant A fragment: row M = qBase+col holds (-2x,-2y,-2z,1)
  float4 qv = qp[qBase + col];
  v2f A;
  A[0] = hi ? qv.z : qv.x;   // K = 2*hi
  A[1] = hi ? qv.w : qv.y;   // K = 2*hi + 1

  float m1[8], m2[8];
  int   i1[8], i2[8];
#pragma unroll
  for (int s = 0; s < 8; ++s) {
    m1[s] = 3.4e38f; m2[s] = 3.4e38f; i1[s] = 0; i2[s] = 0;
  }

#pragma unroll 4
  for (int pBase = 0; pBase < MHALF; pBase += 16) {
    float4 pv = pp[pBase + col];              // column N=col: (x,y,z,p2)
    v2f Bv;
    Bv[0] = hi ? pv.z : pv.x;
    Bv[1] = hi ? pv.w : pv.y;

    v8f c = {};
    c = __builtin_amdgcn_wmma_f32_16x16x4_f32(
        /*neg_a=*/false, A, /*neg_b=*/false, Bv,
        /*c_mod=*/(short)0, c, /*reuse_a=*/false, /*reuse_b=*/false);

    const int pidx = pBase + col;
#pragma unroll
    for (int s = 0; s < 8; ++s) {
      float v   = c[s];
      bool lt1  = v < m1[s];
      bool lt2  = v < m2[s];
      float nm2 = lt1 ? m1[s] : (lt2 ? v : m2[s]);
      int   ni2 = lt1 ? i1[s] : (lt2 ? pidx : i2[s]);
      m1[s] = lt1 ? v : m1[s];
      i1[s] = lt1 ? pidx : i1[s];
      m2[s] = nm2;
      i2[s] = ni2;
    }
  }

  // Butterfly merge of sorted (m1<=m2) pairs across the 16 column-lanes.
  // XOR masks 1..8 stay within each 16-lane half of the wave32.
#pragma unroll
  for (int mask = 8; mask >= 1; mask >>= 1) {
#pragma unroll
    for (int s = 0; s < 8; ++s) {
      float ob1  = __shfl_xor(m1[s], mask, 32);
      int   ob1i = __shfl_xor(i1[s], mask, 32);
      float ob2  = __shfl_xor(m2[s], mask, 32);
      int   ob2i = __shfl_xor(i2[s], mask, 32);

      bool  aF  = m1[s] <= ob1;
      float r1  = aF ? m1[s] : ob1;
      int   r1i = aF ? i1[s] : ob1i;
      float xx  = aF ? ob1 : m1[s];     // max of the two firsts
      int   xxi = aF ? ob1i : i1[s];
      bool  a2s = m2[s] <= ob2;
      float yy  = a2s ? m2[s] : ob2;    // min of the two seconds
      int   yyi = a2s ? i2[s] : ob2i;
      bool  xs  = xx <= yy;

      m1[s] = r1;              i1[s] = r1i;
      m2[s] = xs ? xx : yy;    i2[s] = xs ? xxi : yyi;
    }
  }

  // Lane 0 holds merged rows qBase+0..7 ; lane 16 holds qBase+8..15.
  if (col == 0) {
    const size_t base = (size_t)b * MHALF + qBase + hi * 8;
#pragma unroll
    for (int s = 0; s < 8; ++s) {
      nidx_out[base + s] = (float)i2[s];    // index of 2nd-nearest valid point
    }
  }
}

// ---------------------------------------------------------------------------
extern "C" void kernel_launch(void* const* d_in, const int* in_sizes, int n_in,
                              void* d_out, int out_size, void* d_ws, size_t ws_size,
                              hipStream_t stream) {
  const float* pc    = (const float*)d_in[0];   // (B, N, 3)  f32
  const float* feats = (const float*)d_in[1];   // (B, N, 64) f32
  const int*   perm  = (const int*)d_in[2];     // (N,)       i32
  float* out = (float*)d_out;

  // Workspace: ppts (B*8192 float4) then qpts (B*8192 float4) = 1 MiB total.
  float4* ppts = (float4*)d_ws;
  float4* qpts = ppts + (size_t)BATCH * MHALF;

  {
    int threads = BATCH * MHALF;                    // 32768
    gather_pts_kernel<<<threads / 256, 256, 0, stream>>>(pc, perm, out, ppts, qpts);
  }
  {
    int threads = BATCH * MHALF * FD4;              // 524288
    gather_feats_kernel<<<threads / 256, 256, 0, stream>>>(
        (const float4*)feats, perm, (float4*)(out + VF_OFF));
  }
  {
    int threads = (BATCH * (MHALF / 16)) * 32;      // 2048 waves -> 65536
    knn2_wmma_kernel<<<threads / 256, 256, 0, stream>>>(qpts, ppts, out + NIDX_OFF);
  }
}